// BlurModel_83640193122956
// MI455X (gfx1250) — compile-verified
//
#include <hip/hip_runtime.h>
#include <math.h>

// ---------------------------------------------------------------------------
// BlurModel pipeline for MI455X (gfx1250, wave32):
//   1) zero global histogram (2048 bins aligned to the 0.0005 threshold grid)
//   2) blur kernel: separable 5x5 box; horizontal 5-sum in LDS, vertical
//      5-sum done as banded-matmul via V_WMMA_F32_16X16X4_F32 (5 K=4 chunks);
//      blurred plane -> d_out, histogram -> LDS -> global atomics
//   3) tiny scalar kernel: suffix sums + the two while-loops -> th
//   4) dilate: dilated(u8) = (maxpool5(blur) > th), separable max in LDS
//   5) erode : out(f32)   = minpool5(dilated), separable min in LDS
// Memory-bound: ~576 MB total traffic => ~25us at 23.3 TB/s.
// ---------------------------------------------------------------------------

typedef __attribute__((ext_vector_type(2))) float v2f;
typedef __attribute__((ext_vector_type(8))) float v8f;

#define IMG_H   1024
#define IMG_W   1024
#define PLANES  32               // 8 * 4
#define NTOT    33554432.0f      // 8*4*1024*1024
#define NBINS   2048             // bin width 0.0005, aligned to threshold grid
#define TILE    32               // 32x32 output tile per workgroup
#define TPP     1024             // tiles per plane (32x32 grid of tiles)

__global__ void zero_hist_kernel(unsigned* __restrict__ ghist) {
    int i = blockIdx.x * blockDim.x + threadIdx.x;
    if (i < NBINS) ghist[i] = 0u;
}

// ---- Stage A: blur (WMMA vertical pass) + histogram -----------------------
__global__ __launch_bounds__(128) void blur_wmma_hist_kernel(
    const float* __restrict__ x, const float* __restrict__ kw_ptr,
    float* __restrict__ blur, unsigned* __restrict__ ghist) {

    __shared__ float    s_in[36 * 37];   // input tile + 2-halo, padded stride
    __shared__ float    s_hs[36 * 33];   // horizontal 5-sums, rows -2..33
    __shared__ unsigned s_hist[NBINS];

    const int tid   = threadIdx.x;
    const int bx    = blockIdx.x;
    const int plane = bx >> 10;                  // 1024 tiles per plane
    const int t     = bx & (TPP - 1);
    const int tr    = (t >> 5) << 5;             // tile row origin
    const int tc    = (t & 31) << 5;             // tile col origin

    const float* xp = x    + (size_t)plane * (IMG_H * IMG_W);
    float*       bp = blur + (size_t)plane * (IMG_H * IMG_W);
    const float  kw = kw_ptr[0];                 // 1/25 (all taps equal)

    // load 36x36 input tile (zero pad == SAME zero padding of the conv)
    for (int i = tid; i < 36 * 36; i += 128) {
        int r = i / 36, c = i - r * 36;
        int gr = tr + r - 2, gc = tc + c - 2;
        float v = 0.0f;
        if ((unsigned)gr < IMG_H && (unsigned)gc < IMG_W) v = xp[gr * IMG_W + gc];
        s_in[r * 37 + c] = v;
    }
    for (int i = tid; i < NBINS; i += 128) s_hist[i] = 0u;
    __syncthreads();

    // horizontal 5-sum: rows -2..33 (36), output cols 0..31
    for (int i = tid; i < 36 * 32; i += 128) {
        int hr = i >> 5, c = i & 31;
        const float* p = &s_in[hr * 37 + c];
        s_hs[hr * 33 + c] = p[0] + p[1] + p[2] + p[3] + p[4];
    }
    __syncthreads();

    // vertical 5-sum as banded matmul: Out(16x16) = Band(16x20) x Hsum(20x16)
    // five accumulating V_WMMA_F32_16X16X4_F32 ops, Band generated in-register.
    const int lane = tid & 31;
    const int wave = tid >> 5;                // 4 waves: 2x2 16x16 subtiles
    const int r0   = (wave >> 1) << 4;
    const int c0   = (wave & 1) << 4;
    const int m    = lane & 15;               // A: row / B,C,D: column
    const int hi   = lane >> 4;               // half-wave select

    v8f acc = {};
#pragma unroll
    for (int ch = 0; ch < 5; ++ch) {
        // A (16x4 f32 layout): VGPR0 <-> K = 4ch + 2*hi, VGPR1 <-> +1
        int kk0 = 4 * ch + (hi << 1);
        float a0 = (kk0     >= m && kk0     <= m + 4) ? kw : 0.0f;
        float a1 = (kk0 + 1 >= m && kk0 + 1 <= m + 4) ? kw : 0.0f;
        v2f A = {a0, a1};
        // B (4x16 f32 layout, mirrored): lane half selects K pair, m = column
        int hr = r0 + kk0;                    // hsum row index (= out row -2 +kk +2)
        float b0 = s_hs[hr * 33 + c0 + m];
        float b1 = s_hs[(hr + 1) * 33 + c0 + m];
        v2f B = {b0, b1};
        acc = __builtin_amdgcn_wmma_f32_16x16x4_f32(
            false, A, false, B, (short)0, acc, false, false);
    }

    // C/D layout: VGPR i -> M = i + 8*hi, N = m
    const int colg = tc + c0 + m;
#pragma unroll
    for (int i = 0; i < 8; ++i) {
        int rowg = tr + r0 + i + (hi << 3);
        float v  = acc[i];
        bp[rowg * IMG_W + colg] = v;
        int bin = (int)(v * 2000.0f);
        bin = bin < 0 ? 0 : (bin > NBINS - 1 ? NBINS - 1 : bin);
        atomicAdd(&s_hist[bin], 1u);
    }
    __syncthreads();
    for (int i = tid; i < NBINS; i += 128) {
        unsigned c = s_hist[i];
        if (c) atomicAdd(&ghist[i], c);
    }
}

// ---- Stage B: threshold search (replays the two while-loops) --------------
__global__ void th_search_kernel(const unsigned* __restrict__ ghist,
                                 float* __restrict__ gth) {
    __shared__ unsigned sh[NBINS + 1];
    int tid = threadIdx.x;
    for (int i = tid; i < NBINS; i += blockDim.x) sh[i] = ghist[i];
    __syncthreads();
    if (tid == 0) {
        sh[NBINS] = 0u;
        for (int i = NBINS - 1; i >= 0; --i) sh[i] += sh[i + 1]; // suffix counts
        const float invT = 1.0f / NTOT;
        float th = 0.5f;
        for (int g = 0; g < 4000; ++g) {
            int k = (int)floorf(th * 2000.0f + 0.5f);
            k = k < 0 ? 0 : (k > NBINS ? NBINS : k);
            if (((float)sh[k] * invT) < 0.84f) th -= 0.0005f; else break;
        }
        for (int g = 0; g < 4000; ++g) {
            int k = (int)floorf(th * 2000.0f + 0.5f);
            k = k < 0 ? 0 : (k > NBINS ? NBINS : k);
            if (((float)sh[k] * invT) > 0.86f) th += 0.0005f; else break;
        }
        *gth = th;
    }
}

// ---- Stage C1: dilated(u8) = maxpool5(blur) > th --------------------------
__global__ __launch_bounds__(256) void dilate_kernel(
    const float* __restrict__ blur, const float* __restrict__ gth,
    unsigned char* __restrict__ dil) {

    __shared__ float s_b[36 * 37];
    __shared__ float s_h[36 * 33];

    const int tid   = threadIdx.x;
    const int bx    = blockIdx.x;
    const int plane = bx >> 10;
    const int t     = bx & (TPP - 1);
    const int tr    = (t >> 5) << 5;
    const int tc    = (t & 31) << 5;

    const float* bp = blur + (size_t)plane * (IMG_H * IMG_W);
    unsigned char* dp = dil + (size_t)plane * (IMG_H * IMG_W);
    const float th = *gth;

    for (int i = tid; i < 36 * 36; i += 256) {
        int r = i / 36, c = i - r * 36;
        int gr = tr + r - 2, gc = tc + c - 2;
        float v = -1.0e30f;   // -inf pad: clipped max window at borders
        if ((unsigned)gr < IMG_H && (unsigned)gc < IMG_W) v = bp[gr * IMG_W + gc];
        s_b[r * 37 + c] = v;
    }
    __syncthreads();
    for (int i = tid; i < 36 * 32; i += 256) {
        int hr = i >> 5, c = i & 31;
        const float* p = &s_b[hr * 37 + c];
        s_h[hr * 33 + c] = fmaxf(fmaxf(fmaxf(p[0], p[1]), fmaxf(p[2], p[3])), p[4]);
    }
    __syncthreads();
    for (int i = tid; i < 32 * 32; i += 256) {
        int r = i >> 5, c = i & 31;
        const float* p = &s_h[r * 33 + c];
        float mx = fmaxf(fmaxf(fmaxf(p[0], p[33]), fmaxf(p[66], p[99])), p[132]);
        dp[(tr + r) * IMG_W + (tc + c)] = (mx > th) ? (unsigned char)1 : (unsigned char)0;
    }
}

// ---- Stage C2: out(f32) = minpool5(dilated) -------------------------------
__global__ __launch_bounds__(256) void erode_kernel(
    const unsigned char* __restrict__ dil, float* __restrict__ out) {

    __shared__ float s_d[36 * 37];
    __shared__ float s_h[36 * 33];

    const int tid   = threadIdx.x;
    const int bx    = blockIdx.x;
    const int plane = bx >> 10;
    const int t     = bx & (TPP - 1);
    const int tr    = (t >> 5) << 5;
    const int tc    = (t & 31) << 5;

    const unsigned char* dp = dil + (size_t)plane * (IMG_H * IMG_W);
    float* op = out + (size_t)plane * (IMG_H * IMG_W);

    for (int i = tid; i < 36 * 36; i += 256) {
        int r = i / 36, c = i - r * 36;
        int gr = tr + r - 2, gc = tc + c - 2;
        float v = 1.0f;       // +inf-equivalent pad: clipped min window
        if ((unsigned)gr < IMG_H && (unsigned)gc < IMG_W)
            v = (float)dp[gr * IMG_W + gc];
        s_d[r * 37 + c] = v;
    }
    __syncthreads();
    for (int i = tid; i < 36 * 32; i += 256) {
        int hr = i >> 5, c = i & 31;
        const float* p = &s_d[hr * 37 + c];
        s_h[hr * 33 + c] = fminf(fminf(fminf(p[0], p[1]), fminf(p[2], p[3])), p[4]);
    }
    __syncthreads();
    for (int i = tid; i < 32 * 32; i += 256) {
        int r = i >> 5, c = i & 31;
        const float* p = &s_h[r * 33 + c];
        float mn = fminf(fminf(fminf(p[0], p[33]), fminf(p[66], p[99])), p[132]);
        op[(tr + r) * IMG_W + (tc + c)] = mn;   // exactly 0.0 or 1.0
    }
}

// ---------------------------------------------------------------------------
extern "C" void kernel_launch(void* const* d_in, const int* in_sizes, int n_in,
                              void* d_out, int out_size, void* d_ws, size_t ws_size,
                              hipStream_t stream) {
    (void)in_sizes; (void)n_in; (void)out_size; (void)ws_size;

    const float* x  = (const float*)d_in[0];    // (8,4,1024,1024) f32
    const float* kw = (const float*)d_in[1];    // 5x5 box taps (all 1/25)
    float* out = (float*)d_out;                 // blur scratch, then final closed

    // workspace layout: [0,32MB) dilated u8 ; then 2048-u32 hist ; then th
    unsigned char* ws8   = (unsigned char*)d_ws;
    unsigned*      ghist = (unsigned*)((char*)d_ws + (size_t)PLANES * IMG_H * IMG_W);
    float*         gth   = (float*)((char*)ghist + NBINS * sizeof(unsigned));

    const int nblk = PLANES * TPP;  // 32768 tiles

    zero_hist_kernel<<<(NBINS + 255) / 256, 256, 0, stream>>>(ghist);
    blur_wmma_hist_kernel<<<nblk, 128, 0, stream>>>(x, kw, out, ghist);
    th_search_kernel<<<1, 64, 0, stream>>>(ghist, gth);
    dilate_kernel<<<nblk, 256, 0, stream>>>(out, gth, ws8);
    erode_kernel<<<nblk, 256, 0, stream>>>(ws8, out);
}